// AdaptiveSpanAttention_85882166051370
// MI455X (gfx1250) — compile-verified
//
#include <hip/hip_runtime.h>
#include <math.h>

typedef __attribute__((ext_vector_type(2))) float v2f;
typedef __attribute__((ext_vector_type(8))) float v8f;

#define BE 65536  // 64 * 1024 (batch*E)

__device__ __forceinline__ float gelu_exact(float x) {
    return 0.5f * x * (1.0f + erff(x * 0.7071067811865475f));
}
__device__ __forceinline__ float sigmoidf_(float x) {
    return 1.0f / (1.0f + expf(-x));
}

// ---------------- attn fill ----------------
__global__ void fill_kernel(float* __restrict__ out, float v, int n) {
    int i = blockIdx.x * blockDim.x + threadIdx.x;
    if (i < n) out[i] = v;
}

// ---------------- ctx = mean(encoded_x, axis=0), two-stage, deterministic ----------------
__global__ void mean_partial_kernel(const float* __restrict__ enc, float* __restrict__ part) {
    int be = blockIdx.x * blockDim.x + threadIdx.x;   // 0..65535
    int c  = blockIdx.y;                              // 0..15 chunk of 32 seq steps
    const float* p = enc + (size_t)c * 32 * BE + be;
    float s = 0.0f;
    #pragma unroll 4
    for (int l = 0; l < 32; ++l) {
        if (l + 8 < 32) __builtin_prefetch(p + (size_t)(l + 8) * BE, 0, 1);
        s += p[(size_t)l * BE];
    }
    part[(size_t)c * BE + be] = s;
}
__global__ void mean_final_kernel(const float* __restrict__ part, float* __restrict__ ctx) {
    int be = blockIdx.x * blockDim.x + threadIdx.x;
    float s = 0.0f;
    #pragma unroll
    for (int c = 0; c < 16; ++c) s += part[(size_t)c * BE + be];
    ctx[be] = s * (1.0f / 512.0f);
}

// ---------------- gather global_hist = long_x[::128] ----------------
__global__ void gather_kernel(const float* __restrict__ long_x, float* __restrict__ ghist) {
    int idx = blockIdx.x * blockDim.x + threadIdx.x;  // 0..32*65536-1
    int l  = idx >> 16;
    int be = idx & 65535;
    ghist[idx] = long_x[(size_t)l * 128 * BE + be];
}

// ---------------- WMMA f32 GEMM: out = epilogue(A[MxK] @ B[KxN]) ----------------
// One wave per 32(M) x 64(N) macro-tile: 2 A-frags x 4 B-frags -> 8 v_wmma per K-step.
// M multiple of 32, N multiple of 64, macro-tile count multiple of 8 (256-thread blocks).
// MODE 1: gelu(acc + bias[n])
// MODE 2: acc + bias[n]
// MODE 3: gelu(acc + extra[(m%64)*N + n])          (per-batch row add, includes bias)
// MODE 4: sigmoid(acc + bias[n]) * extra[m*N + n]  (gate * fused)
template <int MODE>
__global__ __launch_bounds__(256)
void gemm_wmma(const float* __restrict__ A, int lda,
               const float* __restrict__ B, int ldb,
               const float* __restrict__ bias,
               const float* __restrict__ extra,
               float* __restrict__ out, int ldo,
               int N, int K)
{
    const int wave = threadIdx.x >> 5;
    const int lane = threadIdx.x & 31;
    const int half = lane >> 4;
    const int li   = lane & 15;

    const int nMacroN = N >> 6;                       // macro tiles along N
    const int macro   = blockIdx.x * 8 + wave;        // wave-uniform
    const int tm      = (macro / nMacroN) << 5;
    const int tn      = (macro % nMacroN) << 6;

    v8f acc[2][4];
    #pragma unroll
    for (int mt = 0; mt < 2; ++mt)
        #pragma unroll
        for (int j = 0; j < 4; ++j)
            acc[mt][j] = (v8f){0.f, 0.f, 0.f, 0.f, 0.f, 0.f, 0.f, 0.f};

    // 32-bit A 16x4 frag: lane (li,half) holds A[row=li][k+2*half + {0,1}]
    // 32-bit B 4x16 frag: lane (li,half) holds B[k+2*half + {0,1}][col=li]
    const float* arow0 = A + (size_t)(tm + li) * lda + 2 * half;
    const float* arow1 = arow0 + (size_t)16 * lda;
    const float* bbase = B + (size_t)(2 * half) * ldb + tn + li;

    for (int k = 0; k < K; k += 4) {
        v2f a0 = *(const v2f*)(arow0 + k);
        v2f a1 = *(const v2f*)(arow1 + k);
        v2f b[4];
        #pragma unroll
        for (int j = 0; j < 4; ++j) {
            b[j].x = bbase[(size_t)k * ldb + j * 16];
            b[j].y = bbase[(size_t)(k + 1) * ldb + j * 16];
        }
        #pragma unroll
        for (int j = 0; j < 4; ++j) {
            acc[0][j] = __builtin_amdgcn_wmma_f32_16x16x4_f32(
                false, a0, false, b[j], (short)0, acc[0][j], false, false);
            acc[1][j] = __builtin_amdgcn_wmma_f32_16x16x4_f32(
                false, a1, false, b[j], (short)0, acc[1][j], false, false);
        }
    }

    // C/D layout: VGPR r, lane -> row r + 8*half, col li (within 16x16 tile)
    #pragma unroll
    for (int mt = 0; mt < 2; ++mt) {
        #pragma unroll
        for (int j = 0; j < 4; ++j) {
            const int n = tn + j * 16 + li;
            #pragma unroll
            for (int r = 0; r < 8; ++r) {
                const int m = tm + mt * 16 + 8 * half + r;
                float v = acc[mt][j][r];
                if (MODE == 1)      v = gelu_exact(v + bias[n]);
                else if (MODE == 2) v = v + bias[n];
                else if (MODE == 3) v = gelu_exact(v + extra[(size_t)(m & 63) * N + n]);
                else if (MODE == 4) v = sigmoidf_(v + bias[n]) * extra[(size_t)m * N + n];
                out[(size_t)m * ldo + n] = v;
            }
        }
    }
}

// ---------------- scores = sigmoid(h @ rs_w2 + rs_b2), one wave per row ----------------
__global__ __launch_bounds__(256)
void scores_kernel(const float* __restrict__ h, const float* __restrict__ rs_w2,
                   const float* __restrict__ rs_b2, float* __restrict__ scores)
{
    int wave = threadIdx.x >> 5;
    int lane = threadIdx.x & 31;
    int row  = blockIdx.x * 8 + wave;      // 0..2047
    const float* hr = h + (size_t)row * 512;
    float s = 0.0f;
    #pragma unroll
    for (int k = lane; k < 512; k += 32) s += hr[k] * rs_w2[k];
    #pragma unroll
    for (int off = 16; off > 0; off >>= 1) s += __shfl_down(s, off, 32);
    if (lane == 0) scores[row] = sigmoidf_(s + rs_b2[0]);
}

// ---------------- span_w = softmax(h_sp @ sp_w2 + sp_b2), one wave per batch ----------------
__global__ void span_kernel(const float* __restrict__ h_sp, const float* __restrict__ sp_w2,
                            const float* __restrict__ sp_b2, float* __restrict__ span_w)
{
    int b = blockIdx.x;
    int lane = threadIdx.x;                // 32 threads
    const float* hr = h_sp + (size_t)b * 512;
    float s0 = 0.0f, s1 = 0.0f;
    #pragma unroll
    for (int k = lane; k < 512; k += 32) {
        float hv = hr[k];
        s0 += hv * sp_w2[2 * k];
        s1 += hv * sp_w2[2 * k + 1];
    }
    #pragma unroll
    for (int off = 16; off > 0; off >>= 1) {
        s0 += __shfl_down(s0, off, 32);
        s1 += __shfl_down(s1, off, 32);
    }
    if (lane == 0) {
        s0 += sp_b2[0]; s1 += sp_b2[1];
        float mx = fmaxf(s0, s1);
        float e0 = expf(s0 - mx), e1 = expf(s1 - mx);
        float d = e0 + e1;
        span_w[2 * b]     = e0 / d;
        span_w[2 * b + 1] = e1 / d;
    }
}

// ---------------- w[l][b] = softmax_l(scores + log(decay^(31-l)+1e-8)), wave per batch ----------------
__global__ void softmax_w_kernel(const float* __restrict__ scores, const float* __restrict__ decay_p,
                                 float* __restrict__ w)
{
    int b = blockIdx.x;
    int l = threadIdx.x;                   // 0..31
    float decay = *decay_p;
    float v = scores[l * 64 + b] + logf(powf(decay, (float)(31 - l)) + 1e-8f);
    float m = v;
    #pragma unroll
    for (int off = 16; off > 0; off >>= 1) m = fmaxf(m, __shfl_xor(m, off, 32));
    float e = expf(v - m);
    float s = e;
    #pragma unroll
    for (int off = 16; off > 0; off >>= 1) s += __shfl_xor(s, off, 32);
    w[l * 64 + b] = e / s;
}

// ---------------- combined[b][which*1024 + e] = span_w[b][which] * sum_l w[l][b]*hist[l][b][e] ----------------
__global__ void pool_kernel(const float* __restrict__ hist, const float* __restrict__ w,
                            const float* __restrict__ span_w, int which, float* __restrict__ combined)
{
    int idx = blockIdx.x * blockDim.x + threadIdx.x;  // 0..65535
    int b = idx >> 10;
    int e = idx & 1023;
    float s = 0.0f;
    #pragma unroll 4
    for (int l = 0; l < 32; ++l)
        s += w[l * 64 + b] * hist[((size_t)l * 64 + b) * 1024 + e];
    combined[(size_t)b * 2048 + which * 1024 + e] = s * span_w[2 * b + which];
}

extern "C" void kernel_launch(void* const* d_in, const int* in_sizes, int n_in,
                              void* d_out, int out_size, void* d_ws, size_t ws_size,
                              hipStream_t stream) {
    const float* long_x  = (const float*)d_in[0];
    const float* encoded = (const float*)d_in[1];
    const float* sp_w1   = (const float*)d_in[2];
    const float* sp_b1   = (const float*)d_in[3];
    const float* sp_w2   = (const float*)d_in[4];
    const float* sp_b2   = (const float*)d_in[5];
    const float* rs_w1   = (const float*)d_in[6];
    const float* rs_b1   = (const float*)d_in[7];
    const float* rs_w2   = (const float*)d_in[8];
    const float* rs_b2   = (const float*)d_in[9];
    const float* fu_w    = (const float*)d_in[10];
    const float* fu_b    = (const float*)d_in[11];
    const float* g_w     = (const float*)d_in[12];
    const float* g_b     = (const float*)d_in[13];
    const float* decay_l = (const float*)d_in[14];
    const float* decay_g = (const float*)d_in[15];

    float* out_attn  = (float*)d_out;                 // (4096, 64) = 262144
    float* out_adapt = (float*)d_out + 262144;        // (64, 1024) = 65536

    float* ws       = (float*)d_ws;
    float* ctx      = ws;                   // 65536
    float* part     = ctx + 65536;          // 16*65536 = 1048576
    float* h_sp     = part + 1048576;       // 64*512  = 32768
    float* span_w   = h_sp + 32768;         // 128
    float* ctx_rs   = span_w + 128;         // 64*512  = 32768
    float* ghist    = ctx_rs + 32768;       // 32*65536 = 2097152
    float* h_loc    = ghist + 2097152;      // 2048*512 = 1048576
    float* h_glob   = h_loc + 1048576;      // 1048576
    float* sc_l     = h_glob + 1048576;     // 2048
    float* sc_g     = sc_l + 2048;          // 2048
    float* w_l      = sc_g + 2048;          // 2048
    float* w_g      = w_l + 2048;           // 2048
    float* combined = w_g + 2048;           // 64*2048 = 131072
    float* fused    = combined + 131072;    // 65536
    // total ~5.6M floats (~22.4 MB) of workspace

    const float* lhist = long_x + (size_t)4064 * BE;  // last 32 steps, contiguous (32,64,1024)

    // attn = 1/seq_len everywhere
    fill_kernel<<<1024, 256, 0, stream>>>(out_attn, 1.0f / 4096.0f, 262144);

    // ctx = mean over 512 encoded steps (deterministic two-stage, 128 MB streamed)
    mean_partial_kernel<<<dim3(256, 16), 256, 0, stream>>>(encoded, part);
    mean_final_kernel<<<256, 256, 0, stream>>>(part, ctx);

    // gather strided global history (8 MB) into contiguous ws
    gather_kernel<<<8192, 256, 0, stream>>>(long_x, ghist);

    // span head: h_sp = gelu(ctx @ sp_w1 + sp_b1); span_w = softmax(h_sp @ sp_w2 + sp_b2)
    gemm_wmma<1><<<2, 256, 0, stream>>>(ctx, 1024, sp_w1, 512, sp_b1, nullptr, h_sp, 512, 512, 1024);
    span_kernel<<<64, 32, 0, stream>>>(h_sp, sp_w2, sp_b2, span_w);

    // ctx_rs = ctx @ rs_w1[1024:] + rs_b1  (shared across both pooled calls and all 32 steps)
    gemm_wmma<2><<<2, 256, 0, stream>>>(ctx, 1024, rs_w1 + (size_t)1024 * 512, 512, rs_b1, nullptr,
                                        ctx_rs, 512, 512, 1024);

    // h = gelu(hist @ rs_w1[:1024] + ctx_rs[b])  for local and global histories
    gemm_wmma<3><<<64, 256, 0, stream>>>(lhist, 1024, rs_w1, 512, nullptr, ctx_rs, h_loc, 512, 512, 1024);
    gemm_wmma<3><<<64, 256, 0, stream>>>(ghist, 1024, rs_w1, 512, nullptr, ctx_rs, h_glob, 512, 512, 1024);

    // scores = sigmoid(h @ rs_w2 + rs_b2)
    scores_kernel<<<256, 256, 0, stream>>>(h_loc, rs_w2, rs_b2, sc_l);
    scores_kernel<<<256, 256, 0, stream>>>(h_glob, rs_w2, rs_b2, sc_g);

    // softmax over L with decay prior
    softmax_w_kernel<<<64, 32, 0, stream>>>(sc_l, decay_l, w_l);
    softmax_w_kernel<<<64, 32, 0, stream>>>(sc_g, decay_g, w_g);

    // pooled features, scaled by span_w, packed into combined = [local*s0 | global*s1]
    pool_kernel<<<256, 256, 0, stream>>>(lhist, w_l, span_w, 0, combined);
    pool_kernel<<<256, 256, 0, stream>>>(ghist, w_g, span_w, 1, combined);

    // fused = gelu(combined @ fu_w + fu_b)
    gemm_wmma<1><<<4, 256, 0, stream>>>(combined, 2048, fu_w, 1024, fu_b, nullptr, fused, 1024, 1024, 2048);

    // adaptive = sigmoid(fused @ g_w + g_b) * fused  -> written straight to d_out
    gemm_wmma<4><<<4, 256, 0, stream>>>(fused, 1024, g_w, 1024, g_b, fused, out_adapt, 1024, 1024, 1024);
}